// Yolact_39779987095574
// MI455X (gfx1250) — compile-verified
//
#include <hip/hip_runtime.h>
#include <hip/hip_bf16.h>
#include <math.h>

typedef __attribute__((ext_vector_type(2))) float v2f;
typedef __attribute__((ext_vector_type(8))) float v8f;

#define FPN 256
#define BATCH 8
#define TOTAL_PRIORS 17742
#define OUT_CH 117   /* 4 bbox + 81 conf + 32 mask */

#define WMMA_F32(A, B, C) \
  __builtin_amdgcn_wmma_f32_16x16x4_f32(false, (A), false, (B), (short)0, (C), false, false)

// Opaque register barrier: hides value provenance so the compiler keeps an
// independent 32-bit index register (scale_offset GVS loads) instead of
// re-deriving 64-bit addresses from a sibling index.
__device__ __forceinline__ int opaque_i32(int v) { asm("" : "+v"(v)); return v; }

// ---------------------------------------------------------------------------
// fp32 WMMA 16x16x4. Each wave: 32(Cout) x 64(N) tile, K software-pipelined.
// All B-operand loads use pure 32-bit indices off a uniform base so they
// lower to "global_load_b32 v, vIdx, s[base] scale_offset".
// ---------------------------------------------------------------------------

__global__ void __launch_bounds__(32)
conv1x1_wmma(const float* __restrict__ in, const float* __restrict__ w,
             const float* __restrict__ bias, float* __restrict__ out,
             int Cin, int Cout, int HW)
{
  const int lane = threadIdx.x;
  const int hf   = lane >> 4;
  const int l16  = lane & 15;
  const int n0   = blockIdx.x * 64;
  const int m0   = blockIdx.y * 32;
  const int b    = blockIdx.z;
  const int kr   = hf * 2;

  const float* inb = in + (size_t)b * Cin * HW;

  v8f acc[8];
#pragma unroll
  for (int q = 0; q < 8; q++)
#pragma unroll
    for (int r = 0; r < 8; r++) acc[q][r] = 0.0f;

  int ncol[4]; int nok[4];
#pragma unroll
  for (int t = 0; t < 4; t++) {
    int n = n0 + t * 16 + l16;
    nok[t]  = (n < HW);
    ncol[t] = nok[t] ? n : (HW - 1);   // clamped; garbage cols never stored
  }

  int wiA = (m0 + l16) * Cin + kr;     // even -> v2f (b64) weight loads legal
  int wiB = wiA + 16 * Cin;
  int bi0[4], bi1[4];
#pragma unroll
  for (int t = 0; t < 4; t++) {
    bi0[t] = kr * HW + ncol[t];
    bi1[t] = opaque_i32(bi0[t] + HW);
  }

  const int HW4 = 4 * HW;
  const int kiters = Cin >> 2;

  v2f a0c, a1c, bbc[4];
  a0c = *(const v2f*)(w + wiA);
  a1c = *(const v2f*)(w + wiB);
#pragma unroll
  for (int t = 0; t < 4; t++) { bbc[t].x = inb[bi0[t]]; bbc[t].y = inb[bi1[t]]; }

  for (int it = 1; it < kiters; ++it) {
    wiA += 4; wiB += 4;
#pragma unroll
    for (int t = 0; t < 4; t++) { bi0[t] += HW4; bi1[t] += HW4; }
    v2f a0n, a1n, bbn[4];
    a0n = *(const v2f*)(w + wiA);
    a1n = *(const v2f*)(w + wiB);
#pragma unroll
    for (int t = 0; t < 4; t++) { bbn[t].x = inb[bi0[t]]; bbn[t].y = inb[bi1[t]]; }
#pragma unroll
    for (int t = 0; t < 4; t++) {
      acc[t]     = WMMA_F32(a0c, bbc[t], acc[t]);
      acc[4 + t] = WMMA_F32(a1c, bbc[t], acc[4 + t]);
    }
    a0c = a0n; a1c = a1n;
#pragma unroll
    for (int t = 0; t < 4; t++) bbc[t] = bbn[t];
  }
#pragma unroll
  for (int t = 0; t < 4; t++) {
    acc[t]     = WMMA_F32(a0c, bbc[t], acc[t]);
    acc[4 + t] = WMMA_F32(a1c, bbc[t], acc[4 + t]);
  }

#pragma unroll
  for (int s = 0; s < 2; s++)
#pragma unroll
    for (int t = 0; t < 4; t++) {
      if (!nok[t]) continue;
      int n = ncol[t];
#pragma unroll
      for (int r = 0; r < 8; r++) {
        int m = m0 + s * 16 + r + hf * 8;
        out[((size_t)b * Cout + m) * HW + n] = acc[s * 4 + t][r] + bias[m];
      }
    }
}

template <int PAD>
__global__ void __launch_bounds__(32)
conv3x3_wmma(const float* __restrict__ in, const float* __restrict__ w,
             const float* __restrict__ bias, float* __restrict__ out,
             int Cin, int Cout, int H, int W, int Ho, int Wo, int do_relu)
{
  const int lane = threadIdx.x;
  const int hf   = lane >> 4;
  const int l16  = lane & 15;
  const int n0   = blockIdx.x * 64;
  const int m0   = blockIdx.y * 32;
  const int b    = blockIdx.z;
  const int kr   = hf * 2;
  const int HWo  = Ho * Wo;
  const int HWi  = H * W;
  const int HWi4 = 4 * HWi;

  const float* inb = in + (size_t)b * Cin * HWi;

  v8f acc[8];
#pragma unroll
  for (int q = 0; q < 8; q++)
#pragma unroll
    for (int r = 0; r < 8; r++) acc[q][r] = 0.0f;

  int ox[4], oy[4], nok[4];
#pragma unroll
  for (int t = 0; t < 4; t++) {
    int n = n0 + t * 16 + l16;
    nok[t] = (n < HWo);
    n = nok[t] ? n : 0;
    oy[t] = n / Wo;
    ox[t] = n % Wo;
  }

  const int wrow0 = (m0 + l16) * Cin * 9;
  const int wrow1 = wrow0 + 16 * Cin * 9;
  const int kiters = Cin >> 2;

  for (int kyx = 0; kyx < 9; kyx++) {
    int ky = kyx / 3, kx = kyx % 3;
    int   voff[4];
    float fm[4];
#pragma unroll
    for (int t = 0; t < 4; t++) {
      int iy = oy[t] + ky - PAD;
      int ix = ox[t] + kx - PAD;
      if (PAD) {
        int ok = (iy >= 0) & (iy < H) & (ix >= 0) & (ix < W);
        int iyc = iy < 0 ? 0 : (iy > H - 1 ? H - 1 : iy);
        int ixc = ix < 0 ? 0 : (ix > W - 1 ? W - 1 : ix);
        voff[t] = iyc * W + ixc;
        fm[t]   = ok ? 1.0f : 0.0f;
      } else {
        voff[t] = iy * W + ix;   // pad=0: always in-bounds for valid n
        fm[t]   = 1.0f;
      }
    }

    int wiA = wrow0 + kyx + kr * 9;
    int wiB = wrow1 + kyx + kr * 9;
    int bi0[4], bi1[4];
#pragma unroll
    for (int t = 0; t < 4; t++) {
      bi0[t] = kr * HWi + voff[t];
      bi1[t] = opaque_i32(bi0[t] + HWi);
    }

    v2f a0c, a1c, bbc[4];
    a0c.x = w[wiA]; a0c.y = w[wiA + 9];
    a1c.x = w[wiB]; a1c.y = w[wiB + 9];
#pragma unroll
    for (int t = 0; t < 4; t++) {
      float bx = inb[bi0[t]], by = inb[bi1[t]];
      bbc[t].x = PAD ? bx * fm[t] : bx;
      bbc[t].y = PAD ? by * fm[t] : by;
    }

    for (int it = 1; it < kiters; ++it) {
      wiA += 36; wiB += 36;
#pragma unroll
      for (int t = 0; t < 4; t++) { bi0[t] += HWi4; bi1[t] += HWi4; }
      v2f a0n, a1n, bbn[4];
      a0n.x = w[wiA]; a0n.y = w[wiA + 9];
      a1n.x = w[wiB]; a1n.y = w[wiB + 9];
#pragma unroll
      for (int t = 0; t < 4; t++) {
        float bx = inb[bi0[t]], by = inb[bi1[t]];
        bbn[t].x = PAD ? bx * fm[t] : bx;
        bbn[t].y = PAD ? by * fm[t] : by;
      }
#pragma unroll
      for (int t = 0; t < 4; t++) {
        acc[t]     = WMMA_F32(a0c, bbc[t], acc[t]);
        acc[4 + t] = WMMA_F32(a1c, bbc[t], acc[4 + t]);
      }
      a0c = a0n; a1c = a1n;
#pragma unroll
      for (int t = 0; t < 4; t++) bbc[t] = bbn[t];
    }
#pragma unroll
    for (int t = 0; t < 4; t++) {
      acc[t]     = WMMA_F32(a0c, bbc[t], acc[t]);
      acc[4 + t] = WMMA_F32(a1c, bbc[t], acc[4 + t]);
    }
  }

#pragma unroll
  for (int s = 0; s < 2; s++)
#pragma unroll
    for (int t = 0; t < 4; t++) {
      if (!nok[t]) continue;
      int n = oy[t] * Wo + ox[t];
#pragma unroll
      for (int r = 0; r < 8; r++) {
        int m = m0 + s * 16 + r + hf * 8;
        float v = acc[s * 4 + t][r] + bias[m];
        if (do_relu) v = v > 0.0f ? v : 0.0f;
        out[((size_t)b * Cout + m) * HWo + n] = v;
      }
    }
}

// Head conv: 3x3 pad=1, Cin=FPN, 16(Cout) x 64(N) tile, writes directly into
// (B, TOTAL_PRIORS, 117) with NHWC-transpose + reshape + concat fused.
__global__ void __launch_bounds__(32)
head_conv_wmma(const float* __restrict__ in, const float* __restrict__ w,
               const float* __restrict__ bias, float* __restrict__ out,
               int Cout, int S, int ncomp, int chbase, int prior_off, int do_tanh)
{
  const int lane = threadIdx.x;
  const int hf   = lane >> 4;
  const int l16  = lane & 15;
  const int n0   = blockIdx.x * 64;
  const int m0   = blockIdx.y * 16;
  const int b    = blockIdx.z;
  const int kr   = hf * 2;
  const int HW   = S * S;
  const int HW4  = 4 * HW;

  const float* inb = in + (size_t)b * FPN * HW;

  v8f acc[4];
#pragma unroll
  for (int t = 0; t < 4; t++)
#pragma unroll
    for (int r = 0; r < 8; r++) acc[t][r] = 0.0f;

  int ox[4], oy[4], nok[4];
#pragma unroll
  for (int t = 0; t < 4; t++) {
    int n = n0 + t * 16 + l16;
    nok[t] = (n < HW);
    n = nok[t] ? n : 0;
    oy[t] = n / S;
    ox[t] = n % S;
  }

  int mA = m0 + l16;                 // clamp A row; garbage rows never stored
  if (mA >= Cout) mA = Cout - 1;
  const int wrow = mA * FPN * 9;

  for (int kyx = 0; kyx < 9; kyx++) {
    int ky = kyx / 3, kx = kyx % 3;
    int   voff[4];
    float fm[4];
#pragma unroll
    for (int t = 0; t < 4; t++) {
      int iy = oy[t] + ky - 1;
      int ix = ox[t] + kx - 1;
      int ok = (iy >= 0) & (iy < S) & (ix >= 0) & (ix < S);
      int iyc = iy < 0 ? 0 : (iy > S - 1 ? S - 1 : iy);
      int ixc = ix < 0 ? 0 : (ix > S - 1 ? S - 1 : ix);
      voff[t] = iyc * S + ixc;
      fm[t]   = ok ? 1.0f : 0.0f;
    }

    int wi = wrow + kyx + kr * 9;
    int bi0[4], bi1[4];
#pragma unroll
    for (int t = 0; t < 4; t++) {
      bi0[t] = kr * HW + voff[t];
      bi1[t] = opaque_i32(bi0[t] + HW);
    }

    v2f ac, bbc[4];
    ac.x = w[wi]; ac.y = w[wi + 9];
#pragma unroll
    for (int t = 0; t < 4; t++) {
      bbc[t].x = inb[bi0[t]] * fm[t];
      bbc[t].y = inb[bi1[t]] * fm[t];
    }
    for (int it = 1; it < (FPN >> 2); ++it) {
      wi += 36;
#pragma unroll
      for (int t = 0; t < 4; t++) { bi0[t] += HW4; bi1[t] += HW4; }
      v2f an, bbn[4];
      an.x = w[wi]; an.y = w[wi + 9];
#pragma unroll
      for (int t = 0; t < 4; t++) {
        bbn[t].x = inb[bi0[t]] * fm[t];
        bbn[t].y = inb[bi1[t]] * fm[t];
      }
#pragma unroll
      for (int t = 0; t < 4; t++) acc[t] = WMMA_F32(ac, bbc[t], acc[t]);
      ac = an;
#pragma unroll
      for (int t = 0; t < 4; t++) bbc[t] = bbn[t];
    }
#pragma unroll
    for (int t = 0; t < 4; t++) acc[t] = WMMA_F32(ac, bbc[t], acc[t]);
  }

#pragma unroll
  for (int t = 0; t < 4; t++) {
    if (!nok[t]) continue;
#pragma unroll
    for (int r = 0; r < 8; r++) {
      int m = m0 + r + hf * 8;
      if (m >= Cout) continue;
      int pidx = m / ncomp;
      int c    = m % ncomp;
      size_t prior = (size_t)prior_off + (size_t)(oy[t] * S + ox[t]) * 3 + pidx;
      float v = acc[t][r] + bias[m];
      if (do_tanh) v = tanhf(v);
      out[((size_t)b * TOTAL_PRIORS + prior) * OUT_CH + chbase + c] = v;
    }
  }
}

// dst += bilinear_resize(src)  (jax.image.resize half-pixel semantics)
__global__ void resize_add_kernel(const float* __restrict__ src, float* __restrict__ dst,
                                  int Hin, int Win, int Hout, int Wout, int BC)
{
  int idx = blockIdx.x * blockDim.x + threadIdx.x;
  int total = BC * Hout * Wout;
  if (idx >= total) return;
  int x  = idx % Wout;
  int t  = idx / Wout;
  int y  = t % Hout;
  int bc = t / Hout;
  float sy = (y + 0.5f) * ((float)Hin / (float)Hout) - 0.5f;
  float sx = (x + 0.5f) * ((float)Win / (float)Wout) - 0.5f;
  int y0 = (int)floorf(sy); float fy = sy - (float)y0;
  int x0 = (int)floorf(sx); float fx = sx - (float)x0;
  int y0c = y0 < 0 ? 0 : (y0 > Hin - 1 ? Hin - 1 : y0);
  int y1 = y0 + 1;
  int y1c = y1 < 0 ? 0 : (y1 > Hin - 1 ? Hin - 1 : y1);
  int x0c = x0 < 0 ? 0 : (x0 > Win - 1 ? Win - 1 : x0);
  int x1 = x0 + 1;
  int x1c = x1 < 0 ? 0 : (x1 > Win - 1 ? Win - 1 : x1);
  const float* s = src + (size_t)bc * Hin * Win;
  float v00 = s[y0c * Win + x0c], v01 = s[y0c * Win + x1c];
  float v10 = s[y1c * Win + x0c], v11 = s[y1c * Win + x1c];
  float v = (1.0f - fy) * ((1.0f - fx) * v00 + fx * v01)
          +          fy * ((1.0f - fx) * v10 + fx * v11);
  dst[idx] += v;
}

// out[bc, y, x] = in[bc, 2y, 2x]
__global__ void subsample_kernel(const float* __restrict__ in, float* __restrict__ out,
                                 int Hin, int Win, int Hout, int Wout, int BC)
{
  int idx = blockIdx.x * blockDim.x + threadIdx.x;
  int total = BC * Hout * Wout;
  if (idx >= total) return;
  int x  = idx % Wout;
  int t  = idx / Wout;
  int y  = t % Hout;
  int bc = t / Hout;
  out[idx] = in[(size_t)bc * Hin * Win + (size_t)(2 * y) * Win + 2 * x];
}

__global__ void priors_kernel(float* __restrict__ pr)
{
  int i = blockIdx.x * blockDim.x + threadIdx.x;
  if (i >= TOTAL_PRIORS) return;
  int S, li;
  if (i < 13467)      { S = 67; li = i; }
  else if (i < 16734) { S = 33; li = i - 13467; }
  else if (i < 17502) { S = 16; li = i - 16734; }
  else if (i < 17694) { S = 8;  li = i - 17502; }
  else                { S = 4;  li = i - 17694; }
  int p = li % 3, cell = li / 3;
  int x = cell % S, y = cell / S;
  float ar = (p == 0) ? 1.0f : (p == 1 ? 0.5f : 2.0f);
  float invS = 1.0f / (float)S;
  pr[i * 4 + 0] = ((float)x + 0.5f) * invS;
  pr[i * 4 + 1] = ((float)y + 0.5f) * invS;
  pr[i * 4 + 2] = 3.0f * ar * invS;
  pr[i * 4 + 3] = 3.0f / ar * invS;
}

static inline int cdiv(int a, int b) { return (a + b - 1) / b; }

extern "C" void kernel_launch(void* const* d_in, const int* in_sizes, int n_in,
                              void* d_out, int out_size, void* d_ws, size_t ws_size,
                              hipStream_t stream)
{
  const float* c3     = (const float*)d_in[0];
  const float* c4     = (const float*)d_in[1];
  const float* c5     = (const float*)d_in[2];
  const float* lat_w5 = (const float*)d_in[3];
  const float* lat_b5 = (const float*)d_in[4];
  const float* lat_w4 = (const float*)d_in[5];
  const float* lat_b4 = (const float*)d_in[6];
  const float* lat_w3 = (const float*)d_in[7];
  const float* lat_b3 = (const float*)d_in[8];
  const float* pred_w0 = (const float*)d_in[9];
  const float* pred_b0 = (const float*)d_in[10];
  const float* pred_w1 = (const float*)d_in[11];
  const float* pred_b1 = (const float*)d_in[12];
  const float* pred_w2 = (const float*)d_in[13];
  const float* pred_b2 = (const float*)d_in[14];
  const float* up_w   = (const float*)d_in[15];
  const float* up_b   = (const float*)d_in[16];
  const float* bbox_w = (const float*)d_in[17];
  const float* bbox_b = (const float*)d_in[18];
  const float* conf_w = (const float*)d_in[19];
  const float* conf_b = (const float*)d_in[20];
  const float* mask_w = (const float*)d_in[21];
  const float* mask_b = (const float*)d_in[22];

  float* out = (float*)d_out;
  float* ws  = (float*)d_ws;

  // workspace layout (floats)
  float* x5 = ws;                       // 8*256*18*18  =   663552
  float* x4 = ws + 663552;              // 8*256*35*35  =  2508800
  float* x3 = ws + 3172352;             // 8*256*69*69  =  9750528
  float* p5 = ws + 12922880;            // 8*256*16*16  =   524288
  float* p4 = ws + 13447168;            // 8*256*33*33  =  2230272
  float* p3 = ws + 15677440;            // 8*256*67*67  =  9193472
  float* p6 = ws + 24868608;            // 8*256*8*8    =   131072
  float* p7 = ws + 24999680;            // 8*256*4*4    =    32768
  float* u  = x3;                       // head scratch aliases dead x3

  dim3 blk(32);

  // x5 = lat5(c5)   (18x18)
  conv1x1_wmma<<<dim3(cdiv(18*18, 64), 8, BATCH), blk, 0, stream>>>(
      c5, lat_w5, lat_b5, x5, 2048, FPN, 18*18);
  // p5 = relu(conv3x3 pad0) : 18 -> 16
  conv3x3_wmma<0><<<dim3(cdiv(16*16, 64), 8, BATCH), blk, 0, stream>>>(
      x5, pred_w0, pred_b0, p5, FPN, FPN, 18, 18, 16, 16, 1);

  // x4 = lat4(c4) + resize(x5, 18->35)
  conv1x1_wmma<<<dim3(cdiv(35*35, 64), 8, BATCH), blk, 0, stream>>>(
      c4, lat_w4, lat_b4, x4, 1024, FPN, 35*35);
  resize_add_kernel<<<cdiv(BATCH*FPN*35*35, 256), 256, 0, stream>>>(
      x5, x4, 18, 18, 35, 35, BATCH*FPN);
  // p4 : 35 -> 33
  conv3x3_wmma<0><<<dim3(cdiv(33*33, 64), 8, BATCH), blk, 0, stream>>>(
      x4, pred_w1, pred_b1, p4, FPN, FPN, 35, 35, 33, 33, 1);

  // x3 = lat3(c3) + resize(x4, 35->69)
  conv1x1_wmma<<<dim3(cdiv(69*69, 64), 8, BATCH), blk, 0, stream>>>(
      c3, lat_w3, lat_b3, x3, 512, FPN, 69*69);
  resize_add_kernel<<<cdiv(BATCH*FPN*69*69, 256), 256, 0, stream>>>(
      x4, x3, 35, 35, 69, 69, BATCH*FPN);
  // p3 : 69 -> 67
  conv3x3_wmma<0><<<dim3(cdiv(67*67, 64), 8, BATCH), blk, 0, stream>>>(
      x3, pred_w2, pred_b2, p3, FPN, FPN, 69, 69, 67, 67, 1);

  // p6 = p5[::2,::2], p7 = p6[::2,::2]
  subsample_kernel<<<cdiv(BATCH*FPN*8*8, 256), 256, 0, stream>>>(
      p5, p6, 16, 16, 8, 8, BATCH*FPN);
  subsample_kernel<<<cdiv(BATCH*FPN*4*4, 256), 256, 0, stream>>>(
      p6, p7, 8, 8, 4, 4, BATCH*FPN);

  // prediction head over 5 levels (shared weights), stores fused into d_out
  const float* lvl_p[5]   = { p3, p4, p5, p6, p7 };
  const int    lvl_S[5]   = { 67, 33, 16, 8, 4 };
  const int    lvl_off[5] = { 0, 13467, 16734, 17502, 17694 };

  for (int L = 0; L < 5; L++) {
    int S = lvl_S[L];
    int HW = S * S;
    int poff = lvl_off[L];
    // u = relu(conv3x3 pad1 (up))
    conv3x3_wmma<1><<<dim3(cdiv(HW, 64), 8, BATCH), blk, 0, stream>>>(
        lvl_p[L], up_w, up_b, u, FPN, FPN, S, S, S, S, 1);
    // bbox: Cout=12, 4 comps, ch [0,4)
    head_conv_wmma<<<dim3(cdiv(HW, 64), cdiv(12, 16), BATCH), blk, 0, stream>>>(
        u, bbox_w, bbox_b, out, 12, S, 4, 0, poff, 0);
    // conf: Cout=243, 81 comps, ch [4,85)
    head_conv_wmma<<<dim3(cdiv(HW, 64), cdiv(243, 16), BATCH), blk, 0, stream>>>(
        u, conf_w, conf_b, out, 243, S, 81, 4, poff, 0);
    // mask: Cout=96, 32 comps, ch [85,117), tanh
    head_conv_wmma<<<dim3(cdiv(HW, 64), cdiv(96, 16), BATCH), blk, 0, stream>>>(
        u, mask_w, mask_b, out, 96, S, 32, 85, poff, 1);
  }

  // priors appended after (B, TOTAL, 117)
  priors_kernel<<<cdiv(TOTAL_PRIORS, 256), 256, 0, stream>>>(
      out + (size_t)BATCH * TOTAL_PRIORS * OUT_CH);
}